// Method_GCN_Class_27032524161531
// MI455X (gfx1250) — compile-verified
//
#include <hip/hip_runtime.h>
#include <cmath>

typedef float v2f __attribute__((ext_vector_type(2)));
typedef float v8f __attribute__((ext_vector_type(8)));

#define NN   100000
#define EE   3200000
#define INF_ 1433
#define HID  16
#define OUTC 7

// ---------------------------------------------------------------- utilities
__global__ void zero_f32(float* __restrict__ p, int n) {
    int i = blockIdx.x * blockDim.x + threadIdx.x;
    if (i < n) p[i] = 0.0f;
}

// ------------------------------------------------- GEMM1: h1 = x @ W1 (WMMA)
// One wave per 16-row tile. A = 16x4 tile of x, B = 4x16 tile of W1,
// C/D = 16x16 fp32 accumulator, K = 1433 in steps of 4.
__global__ __launch_bounds__(256) void gemm1_wmma(
    const float* __restrict__ x, const float* __restrict__ W1,
    float* __restrict__ h1) {
    int wave  = (blockIdx.x * blockDim.x + threadIdx.x) >> 5;
    int lane  = threadIdx.x & 31;
    if (wave >= NN / 16) return;          // uniform per-wave exit (EXEC stays all-1)
    int idx   = lane & 15;                // row of A / col of B / N of C
    int khalf = lane >> 4;                // which K-pair this lane holds
    int r0    = wave * 16;

    const float* xrow = x + (size_t)(r0 + idx) * INF_;
    v8f c = {};

    // main loop: 358 full K-steps of 4 (kb+3 <= 1431 < 1433, no guards)
    for (int kb = 0; kb < 1432; kb += 4) {
        int ka = kb + 2 * khalf;
        v2f a, b;
        a.x = xrow[ka];
        a.y = xrow[ka + 1];
        b.x = W1[ka * HID + idx];
        b.y = W1[(ka + 1) * HID + idx];
        c = __builtin_amdgcn_wmma_f32_16x16x4_f32(
                false, a, false, b, (short)0, c, false, false);
    }
    // tail: K = 1432 (only 1 valid element of the 4-step)
    {
        int ka = 1432 + 2 * khalf;
        v2f a, b;
        a.x = (ka     < INF_) ? xrow[ka]               : 0.0f;
        a.y = (ka + 1 < INF_) ? xrow[ka + 1]           : 0.0f;
        b.x = (ka     < INF_) ? W1[ka * HID + idx]     : 0.0f;
        b.y = (ka + 1 < INF_) ? W1[(ka + 1) * HID + idx] : 0.0f;
        c = __builtin_amdgcn_wmma_f32_16x16x4_f32(
                false, a, false, b, (short)0, c, false, false);
    }
    // store C: VGPR v, lane l -> M = v + 8*(l>>4), N = l&15
    float* out = h1 + (size_t)r0 * HID;
#pragma unroll
    for (int v = 0; v < 8; ++v) {
        int m = v + 8 * khalf;
        out[m * HID + idx] = c[v];
    }
}

// --------------------------------------- SpMM1: acc1[r][j] += val * h1[c][j]
// 16 threads per edge (j = tid&15) -> coalesced atomics per column group.
__global__ __launch_bounds__(256) void spmm1_atomic(
    const int* __restrict__ arow, const int* __restrict__ acol,
    const float* __restrict__ aval, const float* __restrict__ h1,
    float* __restrict__ acc1) {
    int t = blockIdx.x * blockDim.x + threadIdx.x;
    if (t >= EE * 16) return;
    int j = t & 15;
    int e = t >> 4;
    int r = arow[e];
    int ccol = acol[e];
    float v = aval[e] * h1[(size_t)ccol * HID + j];
    atomicAdd(&acc1[(size_t)r * HID + j], v);
}

// ------------------- GEMM2: h2 = relu(acc1 + b1) @ W2  (bias+ReLU fused, WMMA)
// K = 16 (4 WMMA steps), W2 padded from 7 to 16 columns with zeros.
__global__ __launch_bounds__(256) void gemm2_wmma(
    const float* __restrict__ acc1, const float* __restrict__ b1,
    const float* __restrict__ W2, float* __restrict__ h2) {
    int wave  = (blockIdx.x * blockDim.x + threadIdx.x) >> 5;
    int lane  = threadIdx.x & 31;
    if (wave >= NN / 16) return;
    int idx   = lane & 15;
    int khalf = lane >> 4;
    int r0    = wave * 16;

    v8f c = {};
#pragma unroll
    for (int s = 0; s < 4; ++s) {
        int k0 = 4 * s + 2 * khalf;
        v2f a, b;
        float a0 = acc1[(size_t)(r0 + idx) * HID + k0]     + b1[k0];
        float a1 = acc1[(size_t)(r0 + idx) * HID + k0 + 1] + b1[k0 + 1];
        a.x = fmaxf(a0, 0.0f);
        a.y = fmaxf(a1, 0.0f);
        b.x = (idx < OUTC) ? W2[k0 * OUTC + idx]       : 0.0f;
        b.y = (idx < OUTC) ? W2[(k0 + 1) * OUTC + idx] : 0.0f;
        c = __builtin_amdgcn_wmma_f32_16x16x4_f32(
                false, a, false, b, (short)0, c, false, false);
    }
#pragma unroll
    for (int v = 0; v < 8; ++v) {
        int m = v + 8 * khalf;
        if (idx < OUTC) h2[(size_t)(r0 + m) * OUTC + idx] = c[v];
    }
}

// --------------------------------------- SpMM2: acc2[r][j] += val * h2[c][j]
__global__ __launch_bounds__(256) void spmm2_atomic(
    const int* __restrict__ arow, const int* __restrict__ acol,
    const float* __restrict__ aval, const float* __restrict__ h2,
    float* __restrict__ acc2) {
    int t = blockIdx.x * blockDim.x + threadIdx.x;
    if (t >= EE * 8) return;
    int j = t & 7;
    int e = t >> 3;
    if (j >= OUTC) return;
    int r = arow[e];
    int ccol = acol[e];
    float v = aval[e] * h2[(size_t)ccol * OUTC + j];
    atomicAdd(&acc2[(size_t)r * OUTC + j], v);
}

// ------------------------------ final: out = log_softmax(acc2 + b2, axis=1)
__global__ __launch_bounds__(256) void logsoftmax_rows(
    const float* __restrict__ acc2, const float* __restrict__ b2,
    float* __restrict__ out) {
    int r = blockIdx.x * blockDim.x + threadIdx.x;
    if (r >= NN) return;
    float v[OUTC];
    float m = -INFINITY;
#pragma unroll
    for (int j = 0; j < OUTC; ++j) {
        v[j] = acc2[(size_t)r * OUTC + j] + b2[j];
        m = fmaxf(m, v[j]);
    }
    float s = 0.0f;
#pragma unroll
    for (int j = 0; j < OUTC; ++j) s += __expf(v[j] - m);
    float ls = __logf(s);
#pragma unroll
    for (int j = 0; j < OUTC; ++j)
        out[(size_t)r * OUTC + j] = v[j] - m - ls;
}

// --------------------------------------------------------------------------
extern "C" void kernel_launch(void* const* d_in, const int* in_sizes, int n_in,
                              void* d_out, int out_size, void* d_ws, size_t ws_size,
                              hipStream_t stream) {
    const float* x    = (const float*)d_in[0];
    const int*   arow = (const int*)  d_in[1];
    const int*   acol = (const int*)  d_in[2];
    const float* aval = (const float*)d_in[3];
    const float* W1   = (const float*)d_in[4];
    const float* b1   = (const float*)d_in[5];
    const float* W2   = (const float*)d_in[6];
    const float* b2   = (const float*)d_in[7];
    float* out = (float*)d_out;

    // workspace: bufA = h1 (N*16), later reused as h2 (N*7)
    //            bufB = acc1 (N*16), later reused as acc2 (N*7)
    float* bufA = (float*)d_ws;
    float* bufB = bufA + (size_t)NN * HID;

    const int TPB = 256;
    int gemm_blocks = ((NN / 16) * 32 + TPB - 1) / TPB;

    // layer 1
    zero_f32<<<(NN * HID + TPB - 1) / TPB, TPB, 0, stream>>>(bufB, NN * HID);
    gemm1_wmma<<<gemm_blocks, TPB, 0, stream>>>(x, W1, bufA);
    spmm1_atomic<<<(EE * 16 + TPB - 1) / TPB, TPB, 0, stream>>>(
        arow, acol, aval, bufA, bufB);

    // layer 2 (gemm2 reads bufB, writes bufA; then bufB is re-zeroed as acc2)
    gemm2_wmma<<<gemm_blocks, TPB, 0, stream>>>(bufB, b1, W2, bufA);
    zero_f32<<<(NN * OUTC + TPB - 1) / TPB, TPB, 0, stream>>>(bufB, NN * OUTC);
    spmm2_atomic<<<(EE * 8 + TPB - 1) / TPB, TPB, 0, stream>>>(
        arow, acol, aval, bufA, bufB);

    // output
    logsoftmax_rows<<<(NN + TPB - 1) / TPB, TPB, 0, stream>>>(bufB, b2, out);
}